// T_Moduel_57921928953933
// MI455X (gfx1250) — compile-verified
//
#include <hip/hip_runtime.h>

// ---- problem constants (from reference) ----
#define BATCH   2
#define TT      5
#define BT      10          // BATCH*TT
#define CIN     32
#define NH      4
#define DK      8
#define HH      64
#define WW      64
#define HWSZ    4096        // HH*WW
#define NPIX    (BT * HWSZ) // 40960
#define NTILE   (NPIX / 16) // 2560
#define QKV_ELEMS (NH * BT * HWSZ * DK)   // 1,310,720 floats per buffer

typedef __attribute__((ext_vector_type(2))) float v2f;
typedef __attribute__((ext_vector_type(8))) float v8f;

// ---------------------------------------------------------------------------
// Kernel A: QKV projection via fp32 WMMA.
// One wave computes a 16-pixel x 96-channel tile (channels: 0-31 Q, 32-63 K,
// 64-95 V). K=32 reduction done as 8 x v_wmma_f32_16x16x4_f32 per N-tile.
// Output layout: q/k/v[head][bt][pix][8] (32B-contiguous per pixel).
// ---------------------------------------------------------------------------
__global__ __launch_bounds__(256) void qkv_proj_wmma(
    const float* __restrict__ x,
    const float* __restrict__ Wq, const float* __restrict__ Wk,
    const float* __restrict__ Wv,
    float* __restrict__ qb, float* __restrict__ kb, float* __restrict__ vb)
{
  const int wave = threadIdx.x >> 5;
  const int lane = threadIdx.x & 31;
  const int tile = blockIdx.x * 8 + wave;
  if (tile >= NTILE) return;

  const int pix0 = tile * 16;
  const int bt   = pix0 >> 12;       // / HWSZ
  const int p0   = pix0 & (HWSZ - 1);
  const float* xbt = x + (size_t)bt * CIN * HWSZ;

  const int mrow  = lane & 15;          // A: pixel-in-tile ; B: N column
  const int khalf = (lane >> 4) << 1;   // 0 for lanes 0-15, 2 for lanes 16-31

  v8f acc[6] = {};                      // 6 N-tiles of 16 channels

  #pragma unroll
  for (int ks = 0; ks < 8; ks++) {
    const int k0 = ks * 4 + khalf;      // this lane's K pair base
    v2f a;
    a.x = xbt[(size_t)(k0 + 0) * HWSZ + p0 + mrow];
    a.y = xbt[(size_t)(k0 + 1) * HWSZ + p0 + mrow];
    #pragma unroll
    for (int j = 0; j < 6; j++) {
      const int chan = j * 16 + mrow;                       // 0..95
      const float* Ws = (chan < 32) ? Wq : (chan < 64 ? Wk : Wv);
      const int n = chan & 31;                              // row of W (out ch)
      v2f bvec;
      bvec.x = Ws[n * CIN + k0 + 0];
      bvec.y = Ws[n * CIN + k0 + 1];
      acc[j] = __builtin_amdgcn_wmma_f32_16x16x4_f32(
          false, a, false, bvec, (short)0, acc[j], false, false);
    }
  }

  // Scatter D tiles: VGPR r holds pixels p0+r (lanes 0-15) / p0+r+8 (16-31),
  // column = j*16 + (lane&15).
  const int pixadd = (lane >> 4) << 3;
  #pragma unroll
  for (int j = 0; j < 6; j++) {
    const int chan  = j * 16 + mrow;
    const int hc    = chan & 31;
    const int head  = hc >> 3;
    const int c     = hc & 7;
    float* dst = (chan < 32) ? qb : (chan < 64 ? kb : vb);
    float* base = dst + ((size_t)(head * BT + bt) * HWSZ) * DK + c;
    #pragma unroll
    for (int r = 0; r < 8; r++) {
      const int pixel = p0 + r + pixadd;
      base[(size_t)pixel * DK] = acc[j][r];
    }
  }
}

// ---------------------------------------------------------------------------
// Kernel B: dilated local attention, one thread per (head, b, t, y, x).
// Online softmax over all 5 frames x (2r+1)^2 window; OOB positions skipped
// (== -inf mask in reference; center offset always valid).
// Head outputs written channel-planar: ho[bt][head*8+c][pix].
// ---------------------------------------------------------------------------
__global__ __launch_bounds__(256) void local_attn(
    const float* __restrict__ qb, const float* __restrict__ kb,
    const float* __restrict__ vb, float* __restrict__ ho)
{
  const int tid  = blockIdx.x * 256 + threadIdx.x;   // NH*BT*HWSZ = 163840
  const int pix  = tid & (HWSZ - 1);
  const int rest = tid >> 12;
  const int bt   = rest % BT;
  const int head = rest / BT;
  const int b    = bt / TT;
  const int y    = pix >> 6;
  const int xx   = pix & 63;

  const int r = (head < 2) ? 3 : 4;      // RADIUS  = [3,3,4,4]
  const int d = (head & 1) ? 2 : 1;      // DILATION= [1,2,1,2]
  const float scale = 0.35355339059327373f;  // 1/sqrt(8)

  const float* qp = qb + ((size_t)(head * BT + bt) * HWSZ + pix) * DK;
  float q[8];
  #pragma unroll
  for (int c = 0; c < 8; c++) q[c] = qp[c];

  float m = -3.4e38f, l = 0.f;
  float acc[8] = {0.f, 0.f, 0.f, 0.f, 0.f, 0.f, 0.f, 0.f};

  for (int s = 0; s < TT; s++) {
    const float* kbase = kb + (size_t)(head * BT + b * TT + s) * HWSZ * DK;
    const float* vbase = vb + (size_t)(head * BT + b * TT + s) * HWSZ * DK;
    for (int dy = -r; dy <= r; dy++) {
      const int ky = y + dy * d;
      if ((unsigned)ky >= (unsigned)HH) continue;
      for (int dx = -r; dx <= r; dx++) {
        const int kx = xx + dx * d;
        if ((unsigned)kx >= (unsigned)WW) continue;
        const int kp = (ky << 6) + kx;

        const float4* k4 = (const float4*)(kbase + (size_t)kp * DK);
        const float4 ka = k4[0], kb2 = k4[1];
        const float e = (q[0]*ka.x + q[1]*ka.y + q[2]*ka.z + q[3]*ka.w +
                         q[4]*kb2.x + q[5]*kb2.y + q[6]*kb2.z + q[7]*kb2.w) * scale;

        const float4* v4 = (const float4*)(vbase + (size_t)kp * DK);
        const float4 va = v4[0], vb2 = v4[1];

        float wgt;
        if (e > m) {                       // rescale only when max improves
          const float cor = __expf(m - e);
          l *= cor;
          #pragma unroll
          for (int c = 0; c < 8; c++) acc[c] *= cor;
          m = e;
          wgt = 1.f;
        } else {
          wgt = __expf(e - m);
        }
        l += wgt;
        acc[0] += wgt * va.x;  acc[1] += wgt * va.y;
        acc[2] += wgt * va.z;  acc[3] += wgt * va.w;
        acc[4] += wgt * vb2.x; acc[5] += wgt * vb2.y;
        acc[6] += wgt * vb2.z; acc[7] += wgt * vb2.w;
      }
    }
  }

  const float inv = 1.f / l;
  float* out = ho + ((size_t)bt * CIN + head * DK) * HWSZ + pix;
  #pragma unroll
  for (int c = 0; c < 8; c++) out[(size_t)c * HWSZ] = acc[c] * inv;
}

// ---------------------------------------------------------------------------
// Kernel C: output projection (32x32) via fp32 WMMA + fused residual add.
// One wave: 16 pixels x 32 out channels = 2 N-tiles x 8 K-steps.
// ---------------------------------------------------------------------------
__global__ __launch_bounds__(256) void out_proj_wmma(
    const float* __restrict__ ho, const float* __restrict__ Wo,
    const float* __restrict__ x, float* __restrict__ out)
{
  const int wave = threadIdx.x >> 5;
  const int lane = threadIdx.x & 31;
  const int tile = blockIdx.x * 8 + wave;
  if (tile >= NTILE) return;

  const int pix0 = tile * 16;
  const int bt   = pix0 >> 12;
  const int p0   = pix0 & (HWSZ - 1);
  const float* hbt = ho + (size_t)bt * CIN * HWSZ;

  const int mrow  = lane & 15;
  const int khalf = (lane >> 4) << 1;

  v8f acc[2] = {};

  #pragma unroll
  for (int ks = 0; ks < 8; ks++) {
    const int k0 = ks * 4 + khalf;
    v2f a;
    a.x = hbt[(size_t)(k0 + 0) * HWSZ + p0 + mrow];
    a.y = hbt[(size_t)(k0 + 1) * HWSZ + p0 + mrow];
    #pragma unroll
    for (int j = 0; j < 2; j++) {
      const int n = j * 16 + mrow;        // out channel k
      v2f bvec;
      bvec.x = Wo[n * CIN + k0 + 0];
      bvec.y = Wo[n * CIN + k0 + 1];
      acc[j] = __builtin_amdgcn_wmma_f32_16x16x4_f32(
          false, a, false, bvec, (short)0, acc[j], false, false);
    }
  }

  const int pixadd = (lane >> 4) << 3;
  #pragma unroll
  for (int j = 0; j < 2; j++) {
    const int n = j * 16 + mrow;
    const float* xrow = x   + ((size_t)bt * CIN + n) * HWSZ;
    float*       orow = out + ((size_t)bt * CIN + n) * HWSZ;
    #pragma unroll
    for (int r = 0; r < 8; r++) {
      const int pixel = p0 + r + pixadd;
      orow[pixel] = acc[j][r] + xrow[pixel];   // fused residual
    }
  }
}

// ---------------------------------------------------------------------------
extern "C" void kernel_launch(void* const* d_in, const int* in_sizes, int n_in,
                              void* d_out, int out_size, void* d_ws, size_t ws_size,
                              hipStream_t stream) {
  const float* x  = (const float*)d_in[0];
  const float* Wq = (const float*)d_in[1];
  const float* Wk = (const float*)d_in[2];
  const float* Wv = (const float*)d_in[3];
  const float* Wo = (const float*)d_in[4];
  float* out = (float*)d_out;

  float* qb = (float*)d_ws;
  float* kb = qb + QKV_ELEMS;
  float* vb = kb + QKV_ELEMS;
  float* ho = vb + QKV_ELEMS;   // BT*32*HWSZ floats

  // A: 2560 tiles, 8 waves/block -> 320 blocks
  qkv_proj_wmma<<<NTILE / 8, 256, 0, stream>>>(x, Wq, Wk, Wv, qb, kb, vb);
  // B: 163840 threads -> 640 blocks
  local_attn<<<(NH * BT * HWSZ) / 256, 256, 0, stream>>>(qb, kb, vb, ho);
  // C: 2560 tiles -> 320 blocks
  out_proj_wmma<<<NTILE / 8, 256, 0, stream>>>(ho, Wo, x, out);
}